// PointerDecoder_21466246545857
// MI455X (gfx1250) — compile-verified
//
#include <hip/hip_runtime.h>
#include <hip/hip_bf16.h>
#include <math.h>

// Problem constants (fixed by setup_inputs)
#define B_ 32
#define N_ 2048
#define D_ 512
#define E_ 512
#define V_ 32000
#define T_ 32
#define G_ 2048   // 4*D
#define K_ 1024   // E + D
#define NEGBIG (-1e9f)

typedef float v2f __attribute__((ext_vector_type(2)));
typedef float v8f __attribute__((ext_vector_type(8)));
typedef unsigned int u32x4 __attribute__((ext_vector_type(4)));
typedef int i32x4 __attribute__((ext_vector_type(4)));
typedef int i32x8 __attribute__((ext_vector_type(8)));

#if defined(__has_builtin)
#if __has_builtin(__builtin_amdgcn_tensor_load_to_lds) && __has_builtin(__builtin_amdgcn_s_wait_tensorcnt)
#define HAS_TDM 1
#endif
#endif
#ifndef HAS_TDM
#define HAS_TDM 0
#endif

__device__ __forceinline__ float sigmoidf_(float x) { return 1.f / (1.f + expf(-x)); }

#if HAS_TDM
// Raw 32-bit LDS byte offset of a generic pointer that points into LDS.
__device__ __forceinline__ unsigned lds_off_u32(void* p) {
    return (unsigned)(unsigned long long)(__attribute__((address_space(3))) void*)p;
}

// Issue TENSOR_LOAD_TO_LDS of one contiguous row of `nelem` f32 from gsrc -> LDS offset.
// D# per CDNA5 ISA 8.3/8.4: group0 {count=1, lds_addr, global_addr, type=2},
// group1 {data_size=4B, tensor_dim0=nelem, tensor_dim1=1, tile_dim0=nelem, tile_dim1=1,
//         tensor_dim0_stride=nelem}. Groups 2/3 zero (2D tensor).
__device__ __forceinline__ void tdm_load_row_f32(const float* gsrc, unsigned lds_off, int nelem) {
    const unsigned long long ga = (unsigned long long)(uintptr_t)gsrc;
    u32x4 g0;
    g0[0] = 1u;                                                 // count=1 (user descriptor)
    g0[1] = lds_off;                                            // lds_addr (bits 63:32)
    g0[2] = (unsigned)(ga & 0xFFFFFFFFu);                       // global_addr[31:0]
    g0[3] = (unsigned)((ga >> 32) & 0x01FFFFFFu) | (2u << 30);  // addr[56:32] | type=2
    i32x8 g1;
    g1[0] = (int)(2u << 16);                                    // wg_mask=0, data_size=2 (4B)
    g1[1] = (int)(((unsigned)nelem & 0xFFFFu) << 16);           // tensor_dim0[15:0] @ bits 63:48
    g1[2] = (int)((((unsigned)nelem >> 16) & 0xFFFFu) | (1u << 16)); // dim0[31:16] | tensor_dim1=1
    g1[3] = (int)(((unsigned)nelem & 0xFFFFu) << 16);           // tile_dim0 @ bits 127:112
    g1[4] = 1;                                                  // tile_dim1=1, tile_dim2=0
    g1[5] = nelem;                                              // tensor_dim0_stride[31:0]
    g1[6] = 0;                                                  // stride0 hi, stride1 lo (unused)
    g1[7] = 0;
    const i32x4 z4 = {0, 0, 0, 0};
#if __clang_major__ >= 23
    const i32x8 z8 = {0, 0, 0, 0, 0, 0, 0, 0};
    __builtin_amdgcn_tensor_load_to_lds(g0, g1, z4, z4, z8, 0);
#else
    __builtin_amdgcn_tensor_load_to_lds(g0, g1, z4, z4, 0);
#endif
}
#endif

// ---- block reductions (any blockDim multiple of 32, <= 1024) ----
__device__ __forceinline__ float blockReduceSum(float v, float* s_red) {
    const int lane = threadIdx.x & 31, wv = threadIdx.x >> 5;
    const int nw = blockDim.x >> 5;
    for (int o = 16; o > 0; o >>= 1) v += __shfl_xor(v, o, 32);
    if (lane == 0) s_red[wv] = v;
    __syncthreads();
    v = (threadIdx.x < nw) ? s_red[threadIdx.x] : 0.f;
    if (wv == 0) {
        for (int o = 16; o > 0; o >>= 1) v += __shfl_xor(v, o, 32);
        if (lane == 0) s_red[0] = v;
    }
    __syncthreads();
    float r = s_red[0];
    __syncthreads();
    return r;
}

__device__ __forceinline__ float blockReduceMax(float v, float* s_red) {
    const int lane = threadIdx.x & 31, wv = threadIdx.x >> 5;
    const int nw = blockDim.x >> 5;
    for (int o = 16; o > 0; o >>= 1) v = fmaxf(v, __shfl_xor(v, o, 32));
    if (lane == 0) s_red[wv] = v;
    __syncthreads();
    v = (threadIdx.x < nw) ? s_red[threadIdx.x] : -3.4e38f;
    if (wv == 0) {
        for (int o = 16; o > 0; o >>= 1) v = fmaxf(v, __shfl_xor(v, o, 32));
        if (lane == 0) s_red[0] = v;
    }
    __syncthreads();
    float r = s_red[0];
    __syncthreads();
    return r;
}

// ---- h0 = masked mean of encoder states (one-time, HBM-bound 128MB first touch) ----
__global__ void k_init_partial(const float* __restrict__ enc, const int* __restrict__ pad,
                               float* __restrict__ h) {
    const int b = blockIdx.x, tile = blockIdx.y, tid = threadIdx.x;  // 256 thr, 8 tiles
    __shared__ float s_pad[256];
    const int n0 = tile * 256;
    s_pad[tid] = (float)(pad[(size_t)b * N_ + n0 + tid] != 0);
    __syncthreads();
    float acc0 = 0.f, acc1 = 0.f;
    const float* base = enc + ((size_t)b * N_ + n0) * D_;
    for (int n = 0; n < 256; ++n) {
        const float pm = s_pad[n];
        const float* row = base + (size_t)n * D_;
        acc0 += pm * row[tid];
        acc1 += pm * row[tid + 256];
    }
    atomicAdd(&h[b * D_ + tid], acc0);
    atomicAdd(&h[b * D_ + tid + 256], acc1);
}

__global__ void k_init_finalize(const int* __restrict__ pad, float* __restrict__ h,
                                float* __restrict__ c) {
    const int b = blockIdx.x, tid = threadIdx.x;  // 256 thr
    __shared__ float s_red[32];
    float cnt = 0.f;
    for (int n = tid; n < N_; n += 256) cnt += (float)(pad[(size_t)b * N_ + n] != 0);
    cnt = blockReduceSum(cnt, s_red);
    const float invc = 1.f / cnt;
    const float h0 = h[b * D_ + tid] * invc;
    const float h1 = h[b * D_ + tid + 256] * invc;
    h[b * D_ + tid] = h0;        h[b * D_ + tid + 256] = h1;
    c[b * D_ + tid] = h0;        c[b * D_ + tid + 256] = h1;
}

// ---- out[0] = one-hot(sos) (after global memset) ----
__global__ void k_out0(float* __restrict__ out, const int* __restrict__ sos_p) {
    const int b = threadIdx.x;
    if (b < B_) out[(size_t)b * V_ + sos_p[0]] = 1.f;
}

// ---- build XH = [emb[inp] | h]  ([B, 1024]) ----
__global__ void k_prep_x(const float* __restrict__ emb, const int* __restrict__ tseq,
                         const int* __restrict__ sos_p, const float* __restrict__ h,
                         float* __restrict__ XH, int s) {
    const int b = blockIdx.x, tid = threadIdx.x;  // 256 thr
    const int inp = (s == 1) ? sos_p[0] : tseq[(s - 1) * B_ + b];
    const float* er = emb + (size_t)inp * E_;
    float* xr = XH + (size_t)b * K_;
    for (int e = tid; e < E_; e += 256) {
        xr[e] = er[e];
        xr[E_ + e] = h[b * D_ + e];
    }
}

// ---- gates[B,2048] = XH[B,1024] @ [W_ih|W_hh]^T + b_ih + b_hh  via V_WMMA_F32_16X16X4_F32 ----
// One wave per 16x16 output tile; 256 waves (2 M-tiles x 128 N-tiles). K-loop split into
// the W_ih half and the W_hh half so the B-fragment pointer is loop-invariant (no cndmask).
// A frag (16x4 f32): lanes 0-15 hold K=k,k+1 for M=lane; lanes 16-31 hold K=k+2,k+3.
// C/D: vgpr i -> M = i + 8*(lane>=16), N = lane&15.
__global__ void k_gates_wmma(const float* __restrict__ XH, const float* __restrict__ W_ih,
                             const float* __restrict__ W_hh, const float* __restrict__ b_ih,
                             const float* __restrict__ b_hh, float* __restrict__ gates) {
    const int wave = (blockIdx.x * blockDim.x + threadIdx.x) >> 5;  // 0..255
    const int lane = threadIdx.x & 31;
    const int mt = wave >> 7;           // 0..1
    const int nt = wave & 127;          // 0..127
    const int half = lane >> 4, l16 = lane & 15;
    const int row = mt * 16 + l16;      // output row this lane feeds via A
    const int col = nt * 16 + l16;      // output col this lane feeds via B
    const float* arow = XH + (size_t)row * K_ + half * 2;
    const float* bih = W_ih + (size_t)col * E_ + half * 2;  // (k,col) = W_ih[col,k], k contiguous
    const float* bhh = W_hh + (size_t)col * D_ + half * 2;
    v8f acc = {};
    for (int k = 0; k < E_; k += 4) {
        v2f av = *(const v2f*)(arow + k);
        v2f bv = *(const v2f*)(bih + k);
        acc = __builtin_amdgcn_wmma_f32_16x16x4_f32(false, av, false, bv,
                                                    (short)0, acc, false, false);
    }
    const float* arow2 = arow + E_;
    for (int k = 0; k < D_; k += 4) {
        v2f av = *(const v2f*)(arow2 + k);
        v2f bv = *(const v2f*)(bhh + k);
        acc = __builtin_amdgcn_wmma_f32_16x16x4_f32(false, av, false, bv,
                                                    (short)0, acc, false, false);
    }
    const float bias = b_ih[col] + b_hh[col];
#pragma unroll
    for (int i = 0; i < 8; ++i) {
        const int m = mt * 16 + i + 8 * half;
        gates[(size_t)m * G_ + col] = acc[i] + bias;
    }
}

// ---- LSTM activations (gate order i,f,g,o) ----
__global__ void k_lstm_act(const float* __restrict__ gates, float* __restrict__ h,
                           float* __restrict__ c) {
    const int idx = blockIdx.x * blockDim.x + threadIdx.x;  // B*D
    const int b = idx / D_, d = idx - b * D_;
    const float* g = gates + (size_t)b * G_;
    const float ig = sigmoidf_(g[d]);
    const float fg = sigmoidf_(g[D_ + d]);
    const float gg = tanhf(g[2 * D_ + d]);
    const float og = sigmoidf_(g[3 * D_ + d]);
    const float cn = fg * c[idx] + ig * gg;
    const float hn = og * tanhf(cn);
    c[idx] = cn;
    h[idx] = hn;
}

// ---- a[B,512] = h @ W_att^T  via V_WMMA_F32_16X16X4_F32 (64 waves) ----
__global__ void k_att_wmma(const float* __restrict__ h, const float* __restrict__ Watt,
                           float* __restrict__ a) {
    const int wave = (blockIdx.x * blockDim.x + threadIdx.x) >> 5;  // 0..63
    const int lane = threadIdx.x & 31;
    const int mt = wave >> 5;          // 0..1
    const int nt = wave & 31;          // 0..31
    const int half = lane >> 4, l16 = lane & 15;
    const int row = mt * 16 + l16;
    const int col = nt * 16 + l16;
    const float* arow = h + (size_t)row * D_ + half * 2;
    const float* brow = Watt + (size_t)col * D_ + half * 2;  // (h@Watt^T)[r,c]=sum_k h[r,k]Watt[c,k]
    v8f acc = {};
    for (int k = 0; k < D_; k += 4) {
        v2f av = *(const v2f*)(arow + k);
        v2f bv = *(const v2f*)(brow + k);
        acc = __builtin_amdgcn_wmma_f32_16x16x4_f32(false, av, false, bv,
                                                    (short)0, acc, false, false);
    }
#pragma unroll
    for (int i = 0; i < 8; ++i) {
        const int m = mt * 16 + i + 8 * half;
        a[(size_t)m * D_ + col] = acc[i];
    }
}

// ---- scores[b,n] = dot(enc[b,n,:], a[b,:]) + (pointer? 0 : NEG)  (L2-resident stream) ----
// Grid (B, 16): 512 blocks spread the 128 MB L2-hot read across WGPs. The 2 KB query row
// is staged into LDS by the Tensor Data Mover (wave 0 issues TENSOR_LOAD_TO_LDS, waits
// TENSORcnt, block barrier publishes). Dot products stream enc directly with float4 loads,
// wave shuffle reductions, and global_prefetch_b8 covering the next row (32 lanes x 64 B).
__global__ void k_scores_dot(const float* __restrict__ enc, const float* __restrict__ a,
                             const int* __restrict__ pmask, float* __restrict__ scores) {
    const int b = blockIdx.x, tile = blockIdx.y;
    const int tid = threadIdx.x;  // 256 thr = 8 waves
    const int wave = tid >> 5, lane = tid & 31;
    __shared__ float a_s[D_];
#if HAS_TDM
    if (wave == 0) {
        tdm_load_row_f32(a + (size_t)b * D_, lds_off_u32((void*)a_s), D_);
        __builtin_amdgcn_s_wait_tensorcnt(0);
    }
#else
    for (int d = tid; d < D_; d += 256) a_s[d] = a[b * D_ + d];
#endif
    __syncthreads();
    const float4* aq = (const float4*)a_s;
    for (int t = wave; t < 128; t += 8) {
        const int n = tile * 128 + t;
        const float* erow = enc + ((size_t)b * N_ + n) * D_;
        if (t + 8 < 128)
            __builtin_prefetch(erow + 8 * D_ + lane * 16, 0, 3);  // next row, 32x64B strides
        const float4* er = (const float4*)erow;
        float sc = 0.f;
#pragma unroll
        for (int j = 0; j < 4; ++j) {
            const float4 ev = er[lane + 32 * j];
            const float4 av = aq[lane + 32 * j];
            sc += ev.x * av.x + ev.y * av.y + ev.z * av.z + ev.w * av.w;
        }
        for (int o = 16; o > 0; o >>= 1) sc += __shfl_xor(sc, o, 32);
        if (lane == 0)
            scores[(size_t)b * N_ + n] = sc + (pmask[(size_t)b * N_ + n] ? 0.f : NEGBIG);
    }
}

// ---- softmax over N, threshold at 1/n_ptr, scatter-add into vocab row, set eos ----
// Tokens equal to eos are skipped in the scatter: the reference overwrites the eos slot
// afterwards, so skipping removes the atomic/plain-store race while matching semantics.
__global__ void k_softmax_scatter(const float* __restrict__ scores,
                                  const int* __restrict__ pmask,
                                  const int* __restrict__ token_ids,
                                  const int* __restrict__ eos_p,
                                  float* __restrict__ out, int s) {
    const int b = blockIdx.x;
    const int tid = threadIdx.x;  // 512 thr
    __shared__ float s_e[N_];
    __shared__ float s_red[32];
    const float* srow = scores + (size_t)b * N_;
    const int* prow = pmask + (size_t)b * N_;
    float mx = -3.4e38f;
    for (int n = tid; n < N_; n += 512) mx = fmaxf(mx, srow[n]);
    mx = blockReduceMax(mx, s_red);
    float sum = 0.f, cnt = 0.f;
    for (int n = tid; n < N_; n += 512) {
        const float e = expf(srow[n] - mx);
        s_e[n] = e;
        sum += e;
        cnt += (float)(prow[n] != 0);
    }
    sum = blockReduceSum(sum, s_red);
    cnt = blockReduceSum(cnt, s_red);
    const float inv = 1.f / sum;
    const float thr = 1.f / cnt;
    const int eos = eos_p[0];
    float* orow = out + (size_t)s * ((size_t)B_ * V_) + (size_t)b * V_;
    float psum = 0.f;
    for (int n = tid; n < N_; n += 512) {
        const float p = s_e[n] * inv;
        if (prow[n] && p >= thr) {
            psum += p;
            const int tok = token_ids[(size_t)b * N_ + n];
            if (tok != eos) atomicAdd(&orow[tok], p);
        }
    }
    psum = blockReduceSum(psum, s_red);
    if (tid == 0) orow[eos] = 1.f - psum;
}

extern "C" void kernel_launch(void* const* d_in, const int* in_sizes, int n_in,
                              void* d_out, int out_size, void* d_ws, size_t ws_size,
                              hipStream_t stream) {
    (void)in_sizes; (void)n_in; (void)ws_size;
    const float* enc   = (const float*)d_in[0];   // [B,N,D]
    const float* emb   = (const float*)d_in[1];   // [V,E]
    const float* W_ih  = (const float*)d_in[2];   // [4D,E]
    const float* W_hh  = (const float*)d_in[3];   // [4D,D]
    const float* b_ih  = (const float*)d_in[4];   // [4D]
    const float* b_hh  = (const float*)d_in[5];   // [4D]
    const float* W_att = (const float*)d_in[6];   // [D,D]
    const int* pad     = (const int*)d_in[7];     // [B,N]
    const int* pmask   = (const int*)d_in[8];     // [B,N]
    const int* tok     = (const int*)d_in[9];     // [B,N]
    const int* tseq    = (const int*)d_in[10];    // [T,B]
    const int* sos_p   = (const int*)d_in[11];
    const int* eos_p   = (const int*)d_in[12];

    float* out = (float*)d_out;                   // [T,B,V]

    // workspace layout (floats); total ~852 KB
    float* ws     = (float*)d_ws;
    float* h      = ws;                 // B*D
    float* c      = h + B_ * D_;        // B*D
    float* a      = c + B_ * D_;        // B*D
    float* XH     = a + B_ * D_;        // B*K
    float* gates  = XH + B_ * K_;       // B*G
    float* scores = gates + B_ * G_;    // B*N

    // zero output (scatter target) and h accumulator
    hipMemsetAsync(d_out, 0, (size_t)out_size * sizeof(float), stream);
    hipMemsetAsync(h, 0, (size_t)B_ * D_ * sizeof(float), stream);

    k_out0<<<1, 32, 0, stream>>>(out, sos_p);
    k_init_partial<<<dim3(B_, 8), 256, 0, stream>>>(enc, pad, h);
    k_init_finalize<<<B_, 256, 0, stream>>>(pad, h, c);

    for (int s = 1; s < T_; ++s) {
        k_prep_x<<<B_, 256, 0, stream>>>(emb, tseq, sos_p, h, XH, s);
        k_gates_wmma<<<64, 128, 0, stream>>>(XH, W_ih, W_hh, b_ih, b_hh, gates);
        k_lstm_act<<<(B_ * D_) / 256, 256, 0, stream>>>(gates, h, c);
        k_att_wmma<<<16, 128, 0, stream>>>(h, W_att, a);
        k_scores_dot<<<dim3(B_, 16), 256, 0, stream>>>(enc, a, pmask, scores);
        k_softmax_scatter<<<B_, 512, 0, stream>>>(scores, pmask, tok, eos_p, out, s);
    }
}